// ProtT5Conv1dCRF4AAClassifier_84172769068033
// MI455X (gfx1250) — compile-verified
//
#include <hip/hip_runtime.h>
#include <hip/hip_bf16.h>

typedef __attribute__((ext_vector_type(2))) float v2f;
typedef __attribute__((ext_vector_type(8))) float v8f;

#define BB   32
#define LL   1024
#define DD   1024
#define TT   10
#define TP   16      // padded tag dim
#define KC   128     // K-chunk staged in LDS
#define AST  132     // LDS stride for A tile (bank-conflict pad, keeps 16B align)
#define WST  130     // LDS stride for W tile (keeps 8B align)
#define NCH  64      // CRF chunks
#define CS   16      // steps per CRF chunk

// ---------------------------------------------------------------------------
// Kernel 1: Conv1d(T=10,k=3,SAME) as WMMA f32 16x16x4 GEMM.
// Grid: (L/128, B), 256 threads (8 waves). Wave w owns a 16(L)x16(T) tile.
// Double-buffered LDS; embeddings staged with global_load_async_to_lds_b128
// (ASYNCcnt) so global traffic overlaps the WMMA loop.
// emissions stored padded: em[b][l][16] (cols 10..15 are exact zeros).
// ---------------------------------------------------------------------------
__global__ void conv_wmma_kernel(const float* __restrict__ emb,
                                 const float* __restrict__ conv_w,
                                 const float* __restrict__ conv_b,
                                 float* __restrict__ em_out) {
    extern __shared__ float lds[];
    float* sA0 = lds;                         // [130][AST]
    float* sA1 = sA0 + 130 * AST;
    float* sW0 = sA1 + 130 * AST;             // [3][16][WST]
    float* sW1 = sW0 + 3 * TP * WST;

    const int b     = blockIdx.y;
    const int lbase = blockIdx.x * 128;
    const int tid   = threadIdx.x;
    const int wave  = tid >> 5;
    const int lane  = tid & 31;
    const int m     = lane & 15;        // row (A) / col (B,C)
    const int half  = lane >> 4;
    const int koff  = half * 2;         // K sub-offset per ISA 16x4 f32 layout

    // ---- stage embeddings tile: async global->LDS, zero-fill halo rows
    auto stageA = [&](float* sA, int kk0) {
        for (int task = tid; task < 130 * 32; task += 256) {
            int r  = task >> 5;
            int c4 = (task & 31) << 2;
            int l  = lbase + r - 1;
            float* dstp = sA + r * AST + c4;
            if (l >= 0 && l < LL) {
                const float* p = emb + (((size_t)b << 10) + l) * DD + kk0 + c4;
                unsigned loff = (unsigned)(uintptr_t)dstp;   // LDS byte offset
                asm volatile("global_load_async_to_lds_b128 %0, %1, off"
                             :: "v"(loff), "v"(p) : "memory");
            } else {
                *(float4*)dstp = make_float4(0.f, 0.f, 0.f, 0.f);
            }
        }
    };
    // ---- stage weights transposed: sW[k][t][dd] = conv_w[t][kk0+dd][k]
    auto stageW = [&](float* sW, int kk0) {
        for (int task = tid; task < 3 * KC; task += 256) {
            int k  = task / KC;
            int dd = task - k * KC;
#pragma unroll
            for (int t = 0; t < TP; ++t) {
                float w = 0.f;
                if (t < TT) w = conv_w[(size_t)t * (DD * 3) + (size_t)(kk0 + dd) * 3 + k];
                sW[k * (TP * WST) + t * WST + dd] = w;
            }
        }
    };

    float bias = (m < TT) ? conv_b[m] : 0.f;
    v8f acc;
#pragma unroll
    for (int i = 0; i < 8; ++i) acc[i] = bias;

    // prologue: fill buffer 0
    stageA(sA0, 0);
    stageW(sW0, 0);
    asm volatile("s_wait_asynccnt 0x0" ::: "memory");
    __syncthreads();

    const int wrow = wave << 4;
    for (int ch = 0; ch < DD / KC; ++ch) {
        float* curA = (ch & 1) ? sA1 : sA0;
        float* curW = (ch & 1) ? sW1 : sW0;
        if (ch + 1 < DD / KC) {            // kick off next chunk's transfers
            stageA((ch & 1) ? sA0 : sA1, (ch + 1) * KC);
            stageW((ch & 1) ? sW0 : sW1, (ch + 1) * KC);
        }
        // ---- accumulate: 32 K-subchunks of 4, x 3 taps
#pragma unroll 4
        for (int c = 0; c < KC / 4; ++c) {
            const int col = (c << 2) + koff;
#pragma unroll
            for (int k = 0; k < 3; ++k) {
                v2f a, bf;
                const float* pa = curA + (wrow + m + k) * AST + col;     // A[m][K..K+1]
                a[0] = pa[0];  a[1] = pa[1];
                const float* pb = curW + k * (TP * WST) + m * WST + col; // B[K..K+1][n=m]
                bf[0] = pb[0]; bf[1] = pb[1];
                acc = __builtin_amdgcn_wmma_f32_16x16x4_f32(
                        false, a, false, bf, (short)0, acc, false, false);
            }
        }
        asm volatile("s_wait_asynccnt 0x0" ::: "memory");
        __syncthreads();
    }

    // ---- store C tile: lane col = m, rows vi + half*8 (ISA C/D layout)
#pragma unroll
    for (int vi = 0; vi < 8; ++vi) {
        int l = lbase + wrow + vi + half * 8;
        em_out[((((size_t)b << 10) + l) << 4) + m] = acc[vi];
    }
}

// ---------------------------------------------------------------------------
// Kernel 2: CRF chunk products in the log semiring.
// Grid: (NCH, B), 1 wave. Lane j<10 holds column j of the 10x10 chunk matrix.
// P_c = prod_{t in chunk} M_t,  M_t[i][j] = trans[i][j] + em[b,t,j] (mask-gated)
// ---------------------------------------------------------------------------
__global__ void crf_chunk_kernel(const float* __restrict__ em,
                                 const int*  __restrict__ mask,
                                 const float* __restrict__ trans,
                                 float* __restrict__ Pmat) {
    const int c = blockIdx.x;
    const int b = blockIdx.y;
    const int j = threadIdx.x;

    float tj[TT];
#pragma unroll
    for (int i = 0; i < TT; ++i) tj[i] = (j < TT) ? trans[i * TT + j] : -1e30f;

    float P[TT];
#pragma unroll
    for (int i = 0; i < TT; ++i) P[i] = (i == j) ? 0.f : -1e30f;  // identity

    int t0 = 1 + c * CS;
    int t1 = t0 + CS; if (t1 > LL) t1 = LL;
    for (int t = t0; t < t1; ++t) {
        int mk = mask[(b << 10) + t];          // uniform across wave
        if (!mk) continue;                      // masked step == identity
        float emj = (j < TT) ? em[((((size_t)b << 10) + t) << 4) + j] : 0.f;
        float newP[TT];
#pragma unroll
        for (int i = 0; i < TT; ++i) {
            float v[TT];
            float mx = -3e38f;
#pragma unroll
            for (int k = 0; k < TT; ++k) {
                float pik = __shfl(P[i], k, 32);   // row i of P lives across lanes
                v[k] = pik + tj[k];
                mx = fmaxf(mx, v[k]);
            }
            float s = 0.f;
#pragma unroll
            for (int k = 0; k < TT; ++k) s += __expf(v[k] - mx);
            newP[i] = mx + __logf(s) + emj;
        }
#pragma unroll
        for (int i = 0; i < TT; ++i) P[i] = newP[i];
    }
    if (j < TT) {
        float* outp = Pmat + (size_t)(b * NCH + c) * 100;
#pragma unroll
        for (int i = 0; i < TT; ++i) outp[i * TT + j] = P[i];
    }
}

// ---------------------------------------------------------------------------
// Kernel 3: per-batch numerator + alpha0 (x) chunk-matrix chain + denom.
// Grid: B, 1 wave each. atomicAdd(denom - num) into the scalar output.
// ---------------------------------------------------------------------------
__global__ void crf_final_kernel(const float* __restrict__ em,
                                 const int*  __restrict__ mask,
                                 const int*  __restrict__ labels,
                                 const float* __restrict__ start_t,
                                 const float* __restrict__ end_t,
                                 const float* __restrict__ trans,
                                 const float* __restrict__ Pmat,
                                 float* __restrict__ out) {
    const int b = blockIdx.x;
    const int j = threadIdx.x;
    const int base = b << 10;

    // ---- numerator + seq_len, lanes strided over t
    float part = 0.f;
    int   cnt  = 0;
    for (int t = j; t < LL; t += 32) {
        int mk = mask[base + t];
        cnt += mk;
        int yt = labels[base + t];
        float e = em[((size_t)(base + t) << 4) + yt];
        if (t == 0) {
            part += start_t[yt] + e;
        } else if (mk) {
            int yp = labels[base + t - 1];
            part += trans[yp * TT + yt] + e;
        }
    }
#pragma unroll
    for (int o = 16; o > 0; o >>= 1) {
        part += __shfl_xor(part, o, 32);
        cnt  += __shfl_xor(cnt,  o, 32);
    }
    int   last = labels[base + cnt - 1];
    float num  = part + end_t[last];

    // ---- alpha0 through 64 chunk matrices (cheap vector x matrix)
    float alpha = (j < TT) ? (start_t[j] + em[((size_t)base << 4) + j]) : -1e30f;
    for (int c = 0; c < NCH; ++c) {
        const float* P = Pmat + (size_t)(b * NCH + c) * 100;
        float v[TT];
        float mx = -3e38f;
#pragma unroll
        for (int k = 0; k < TT; ++k) {
            float ak  = __shfl(alpha, k, 32);
            float pkj = (j < TT) ? P[k * TT + j] : -1e30f;
            v[k] = ak + pkj;
            mx = fmaxf(mx, v[k]);
        }
        float s = 0.f;
#pragma unroll
        for (int k = 0; k < TT; ++k) s += __expf(v[k] - mx);
        float na = mx + __logf(s);
        alpha = (j < TT) ? na : -1e30f;
    }

    // ---- denom = logsumexp(alpha + end)
    float z  = (j < TT) ? alpha + end_t[j] : -3e38f;
    float mz = z;
#pragma unroll
    for (int o = 16; o > 0; o >>= 1) mz = fmaxf(mz, __shfl_xor(mz, o, 32));
    float se = (j < TT) ? __expf(z - mz) : 0.f;
#pragma unroll
    for (int o = 16; o > 0; o >>= 1) se += __shfl_xor(se, o, 32);

    if (j == 0) {
        float denom = mz + __logf(se);
        atomicAdd(out, denom - num);
    }
}

__global__ void zero_out_kernel(float* out) { out[0] = 0.f; }

// ---------------------------------------------------------------------------
extern "C" void kernel_launch(void* const* d_in, const int* in_sizes, int n_in,
                              void* d_out, int out_size, void* d_ws, size_t ws_size,
                              hipStream_t stream) {
    const float* emb     = (const float*)d_in[0];
    const int*   mask    = (const int*)  d_in[1];
    const int*   labels  = (const int*)  d_in[2];
    const float* conv_w  = (const float*)d_in[3];
    const float* conv_b  = (const float*)d_in[4];
    const float* start_t = (const float*)d_in[5];
    const float* end_t   = (const float*)d_in[6];
    const float* trans   = (const float*)d_in[7];
    float* out = (float*)d_out;

    float* em   = (float*)d_ws;                           // B*L*16 fp32 (2 MB)
    float* Pmat = em + (size_t)BB * LL * TP;              // B*NCH*100 fp32

    zero_out_kernel<<<1, 1, 0, stream>>>(out);

    dim3 gA(LL / 128, BB);
    size_t shmem = (size_t)(2 * 130 * AST + 2 * 3 * TP * WST) * sizeof(float);
    conv_wmma_kernel<<<gA, 256, shmem, stream>>>(emb, conv_w, conv_b, em);

    dim3 gB(NCH, BB);
    crf_chunk_kernel<<<gB, 32, 0, stream>>>(em, mask, trans, Pmat);

    crf_final_kernel<<<BB, 32, 0, stream>>>(em, mask, labels, start_t, end_t,
                                            trans, Pmat, out);
}